// encoder_union_11218454577220
// MI455X (gfx1250) — compile-verified
//
#include <hip/hip_runtime.h>
#include <math.h>

typedef __attribute__((ext_vector_type(2))) float v2f;
typedef __attribute__((ext_vector_type(8))) float v8f;

#define H1d 2
#define H0d 2
#define Dd 32
#define INd 128
#define KH 4        // H1*H0
#define OUTH 4
#define SLOPE 0.2f

__device__ __forceinline__ float leaky(float x) { return x > 0.f ? x : SLOPE * x; }

__device__ __forceinline__ void atomicMaxF(float* addr, float v) {
  // monotonic IEEE bit trick; init value must be -inf
  if (v >= 0.f) atomicMax((int*)addr, __float_as_int(v));
  else          atomicMin((unsigned int*)addr, __float_as_uint(v));
}

// ---------------- w_eff[k][i] = sum_d attn_r[k][d] * W_dst[k*32+d][i] ----------------
__global__ void k_weff(const float* __restrict__ Wdst, const float* __restrict__ attn_r,
                       float* __restrict__ weff) {
  int t = threadIdx.x;           // 256 threads: k = t/128, i = t%128
  int k = t >> 7;
  int i = t & 127;
  float s = 0.f;
  for (int d = 0; d < Dd; ++d)
    s += attn_r[k * Dd + d] * Wdst[(k * Dd + d) * INd + i];
  weff[k * INd + i] = s;
}

// ---------------- init emax=-inf, denom=0, out accumulator = 0 ----------------
__global__ void k_init(float* __restrict__ emax, float* __restrict__ denom,
                       float* __restrict__ outb, int n_ed, long n_out) {
  long t = (long)blockIdx.x * blockDim.x + threadIdx.x;
  long stride = (long)gridDim.x * blockDim.x;
  for (long i = t; i < n_ed; i += stride) { emax[i] = -INFINITY; denom[i] = 0.f; }
  for (long i = t; i < n_out; i += stride) outb[i] = 0.f;
}

// ---------------- feat = A[M x 128] * W^T[128 x 64] via V_WMMA_F32_16X16X4_F32 ----------------
// 4 waves / block, each wave -> 16 rows x 64 cols. W packed in LDS as k-pair float2.
__global__ void __launch_bounds__(128)
k_gemm_feat(const float* __restrict__ A, const float* __restrict__ W,
            float* __restrict__ feat, int Mtot) {
  __shared__ v2f Wp[64 * 64];     // Wp[kp*64 + n] = { W[n][2kp], W[n][2kp+1] }, 32 KB
  for (int idx = threadIdx.x; idx < 64 * 64; idx += 128) {
    int kp = idx >> 6, n = idx & 63;
    v2f w; w.x = W[n * INd + 2 * kp]; w.y = W[n * INd + 2 * kp + 1];
    Wp[idx] = w;
  }
  __syncthreads();

  int wave  = threadIdx.x >> 5;
  int lane  = threadIdx.x & 31;
  int l15   = lane & 15;
  int khalf = lane >> 4;                 // lanes 0-15 -> K pair 0, lanes 16-31 -> K pair 1
  int m0    = blockIdx.x * 64 + wave * 16;
  int mrow  = m0 + l15;
  if (mrow >= Mtot) mrow = Mtot - 1;     // clamped read; stores are guarded
  const float* Arow = A + (long)mrow * INd;

  v8f c0 = {}, c1 = {}, c2 = {}, c3 = {};
  for (int kp = 0; kp < 32; ++kp) {      // K = 128, 4 per WMMA
    v2f a = *(const v2f*)(Arow + kp * 4 + khalf * 2);   // A[m][k..k+1], 8B contiguous
    const v2f* brow = &Wp[(kp * 2 + khalf) * 64];
    v2f b0 = brow[l15];
    v2f b1 = brow[16 + l15];
    v2f b2 = brow[32 + l15];
    v2f b3 = brow[48 + l15];
    c0 = __builtin_amdgcn_wmma_f32_16x16x4_f32(false, a, false, b0, (short)0, c0, false, false);
    c1 = __builtin_amdgcn_wmma_f32_16x16x4_f32(false, a, false, b1, (short)0, c1, false, false);
    c2 = __builtin_amdgcn_wmma_f32_16x16x4_f32(false, a, false, b2, (short)0, c2, false, false);
    c3 = __builtin_amdgcn_wmma_f32_16x16x4_f32(false, a, false, b3, (short)0, c3, false, false);
  }

  // C/D layout: VGPR v, lanes 0-15 -> M=v, lanes 16-31 -> M=v+8; col N = lane&15 (+16*ntile)
  int rbase = m0 + khalf * 8;
#pragma unroll
  for (int v = 0; v < 8; ++v) {
    int row = rbase + v;
    if (row < Mtot) {
      float* o = feat + (long)row * 64 + l15;
      o[0]  = c0[v];
      o[16] = c1[v];
      o[32] = c2[v];
      o[48] = c3[v];
    }
  }
}

// ---------------- el[m*2+h0] = dot(feat[m][h0*32 .. +31], attn_l[h0]) ----------------
__global__ void k_el(const float* __restrict__ feat, const float* __restrict__ attn_l,
                     float* __restrict__ el, int total) {
  int t = blockIdx.x * blockDim.x + threadIdx.x;
  if (t >= total) return;
  int h0 = t & 1;
  long m = t >> 1;
  const float* f = feat + m * 64 + h0 * Dd;
  float s = 0.f;
#pragma unroll
  for (int d = 0; d < Dd; ++d) s += f[d] * attn_l[h0 * Dd + d];
  el[t] = s;
}

// ---------------- er[n][k] = dot(hier0[n], weff[k]) ; one wave per node ----------------
__global__ void k_er(const float* __restrict__ hier0, const float* __restrict__ weff,
                     float* __restrict__ er, int Ndst) {
  int gw   = (int)(((long)blockIdx.x * blockDim.x + threadIdx.x) >> 5);
  int lane = threadIdx.x & 31;
  if (gw >= Ndst) return;
  float4 x  = *(const float4*)(hier0 + (long)gw * INd + lane * 4);
  float4 w0 = *(const float4*)(weff + lane * 4);
  float4 w1 = *(const float4*)(weff + INd + lane * 4);
  float s0 = x.x * w0.x + x.y * w0.y + x.z * w0.z + x.w * w0.w;
  float s1 = x.x * w1.x + x.y * w1.y + x.z * w1.z + x.w * w1.w;
#pragma unroll
  for (int off = 16; off > 0; off >>= 1) {
    s0 += __shfl_down(s0, off, 32);
    s1 += __shfl_down(s1, off, 32);
  }
  if (lane == 0) { er[(long)gw * 2] = s0; er[(long)gw * 2 + 1] = s1; }
}

// ---------------- edge pass 1: segment max of leaky(el[src]+er[dst]) ----------------
__global__ void k_edge_max(const int* __restrict__ src, const int* __restrict__ dst,
                           const float* __restrict__ el, const float* __restrict__ er,
                           float* __restrict__ emax, int E) {
  int e = blockIdx.x * blockDim.x + threadIdx.x;
  if (e >= E) return;
  int s = src[e], d = dst[e];
  float4 l = *(const float4*)(el + (long)s * 4);
  float r0 = er[(long)d * 2], r1 = er[(long)d * 2 + 1];
  float x0 = leaky(l.x + r0), x1 = leaky(l.y + r1);
  float x2 = leaky(l.z + r0), x3 = leaky(l.w + r1);
  float* m = emax + (long)d * 4;
  atomicMaxF(m + 0, x0); atomicMaxF(m + 1, x1);
  atomicMaxF(m + 2, x2); atomicMaxF(m + 3, x3);
}

// ---------------- edge pass 2: denom[d] += exp(x - emax[d]) ----------------
__global__ void k_edge_sum(const int* __restrict__ src, const int* __restrict__ dst,
                           const float* __restrict__ el, const float* __restrict__ er,
                           const float* __restrict__ emax, float* __restrict__ denom, int E) {
  int e = blockIdx.x * blockDim.x + threadIdx.x;
  if (e >= E) return;
  int s = src[e], d = dst[e];
  float4 l  = *(const float4*)(el + (long)s * 4);
  float4 mx = *(const float4*)(emax + (long)d * 4);
  float r0 = er[(long)d * 2], r1 = er[(long)d * 2 + 1];
  float* dn = denom + (long)d * 4;
  atomicAdd(dn + 0, expf(leaky(l.x + r0) - mx.x));
  atomicAdd(dn + 1, expf(leaky(l.y + r1) - mx.y));
  atomicAdd(dn + 2, expf(leaky(l.z + r0) - mx.z));
  atomicAdd(dn + 3, expf(leaky(l.w + r1) - mx.w));
}

// ---------------- edge pass 3: out[d] += alpha * feat_src[s] ; one wave per edge ----------------
__global__ void __launch_bounds__(256)
k_edge_agg(const int* __restrict__ src, const int* __restrict__ dst,
           const float* __restrict__ el, const float* __restrict__ er,
           const float* __restrict__ emax, const float* __restrict__ denom,
           const float* __restrict__ feat, float* __restrict__ out, int E) {
  int gw   = (int)(((long)blockIdx.x * blockDim.x + threadIdx.x) >> 5);
  int lane = threadIdx.x & 31;
  if (gw >= E) return;
  int s = src[gw], d = dst[gw];
  float4 l  = *(const float4*)(el + (long)s * 4);
  float4 mx = *(const float4*)(emax + (long)d * 4);
  float4 dn = *(const float4*)(denom + (long)d * 4);
  float r0 = er[(long)d * 2], r1 = er[(long)d * 2 + 1];
  float xs[4]  = { leaky(l.x + r0), leaky(l.y + r1), leaky(l.z + r0), leaky(l.w + r1) };
  float mxs[4] = { mx.x, mx.y, mx.z, mx.w };
  float dns[4] = { dn.x, dn.y, dn.z, dn.w };
  const float* frow = feat + (long)s * 128;  // [(s*2+h1)*64 + h0*32] == s*128 + j*32
  float* orow = out + (long)d * 128;
#pragma unroll
  for (int j = 0; j < KH; ++j) {
    float alpha = expf(xs[j] - mxs[j]) / dns[j];
    atomicAdd(&orow[j * 32 + lane], alpha * frow[j * 32 + lane]);
  }
}

// ---------------- final: in-place 4x4 softmax-weight head mixing + biases ----------------
__global__ void k_final(const float* __restrict__ fcW, const float* __restrict__ fcb,
                        const float* __restrict__ gbias, float* __restrict__ out, int Ndst) {
  __shared__ float SW[OUTH][KH];
  int tid = threadIdx.x;
  if (tid < OUTH) {
    float w[KH]; float mx = -INFINITY;
#pragma unroll
    for (int k = 0; k < KH; ++k) { w[k] = fcW[tid * KH + k]; mx = fmaxf(mx, w[k]); }
    float sum = 0.f;
#pragma unroll
    for (int k = 0; k < KH; ++k) { w[k] = expf(w[k] - mx); sum += w[k]; }
#pragma unroll
    for (int k = 0; k < KH; ++k) SW[tid][k] = w[k] / sum;
  }
  __syncthreads();
  long t = (long)blockIdx.x * blockDim.x + tid;
  long total = (long)Ndst * Dd;
  if (t >= total) return;
  long n = t >> 5;
  int  dcol = (int)(t & 31);
  float* row = out + n * 128;
  float v[KH];
#pragma unroll
  for (int k = 0; k < KH; ++k) v[k] = row[k * 32 + dcol] + gbias[(k & 1) * Dd + dcol];
  float r[OUTH];
#pragma unroll
  for (int o = 0; o < OUTH; ++o) {
    float s = fcb[o];
#pragma unroll
    for (int k = 0; k < KH; ++k) s += SW[o][k] * v[k];
    r[o] = s;
  }
#pragma unroll
  for (int o = 0; o < OUTH; ++o) row[o * 32 + dcol] = r[o];
}

extern "C" void kernel_launch(void* const* d_in, const int* in_sizes, int n_in,
                              void* d_out, int out_size, void* d_ws, size_t ws_size,
                              hipStream_t stream) {
  const float* hier1  = (const float*)d_in[0];
  const float* hier0  = (const float*)d_in[1];
  const int*   src    = (const int*)d_in[2];
  const int*   dst    = (const int*)d_in[3];
  const float* Wsrc   = (const float*)d_in[4];
  const float* Wdst   = (const float*)d_in[5];
  const float* attn_l = (const float*)d_in[6];
  const float* attn_r = (const float*)d_in[7];
  const float* gbias  = (const float*)d_in[8];
  const float* fcW    = (const float*)d_in[9];
  const float* fcb    = (const float*)d_in[10];
  float* out = (float*)d_out;

  const int NSRC = in_sizes[0] / (H1d * INd);
  const int NDST = in_sizes[1] / INd;
  const int E    = in_sizes[2];
  const int M    = NSRC * H1d;           // rows of the feat_src GEMM

  float* ws    = (float*)d_ws;
  float* feat  = ws;                                 // M * 64
  float* el    = feat  + (size_t)M * 64;             // NSRC * 4
  float* er    = el    + (size_t)NSRC * 4;           // NDST * 2
  float* emax  = er    + (size_t)NDST * 2;           // NDST * 4
  float* denom = emax  + (size_t)NDST * 4;           // NDST * 4
  float* weff  = denom + (size_t)NDST * 4;           // 2 * 128

  k_weff<<<1, 256, 0, stream>>>(Wdst, attn_r, weff);
  k_init<<<2048, 256, 0, stream>>>(emax, denom, out, NDST * 4, (long)NDST * 128);
  k_gemm_feat<<<(M + 63) / 64, 128, 0, stream>>>(hier1, Wsrc, feat, M);
  k_el<<<(NSRC * 4 + 255) / 256, 256, 0, stream>>>(feat, attn_l, el, NSRC * 4);
  k_er<<<(int)(((long)NDST * 32 + 255) / 256), 256, 0, stream>>>(hier0, weff, er, NDST);
  k_edge_max<<<(E + 255) / 256, 256, 0, stream>>>(src, dst, el, er, emax, E);
  k_edge_sum<<<(E + 255) / 256, 256, 0, stream>>>(src, dst, el, er, emax, denom, E);
  k_edge_agg<<<(int)(((long)E * 32 + 255) / 256), 256, 0, stream>>>(src, dst, el, er, emax,
                                                                   denom, feat, out, E);
  k_final<<<(int)(((long)NDST * 32 + 255) / 256), 256, 0, stream>>>(fcW, fcb, gbias, out, NDST);
}